// Hard_score_sample_generator_59055800320096
// MI455X (gfx1250) — compile-verified
//
#include <hip/hip_runtime.h>
#include <hip/hip_bf16.h>
#include <stdint.h>

// Problem constants (from the reference).
constexpr int B = 64;
constexpr int T = 1024;
constexpr int F = 1024;
constexpr int HARD_K     = 10;
constexpr int HARD_K_ABN = 3;
constexpr int CONF_K_NOR = 2;
constexpr int CONF_K_ABN = 10;
constexpr int SLOTS      = 16;   // 1 + 3 + 2 + 10 rows per batch
constexpr int NTHR       = 256;  // 8 wave32 waves per workgroup
constexpr int NWAVES     = NTHR / 32;

// ---------------------------------------------------------------------------
// Block-wide argmax over key[0..T-1] with jax.lax.top_k tie-breaking
// (equal values -> lower index wins). wave32 shfl_xor butterfly for the
// intra-wave part (no barrier), tiny LDS merge across the 8 waves
// (2 barriers total). Marks the winner taken (-inf) so repeated calls yield
// the descending top-k sequence. Returns the index uniformly to all threads.
// ---------------------------------------------------------------------------
__device__ __forceinline__ int block_argmax_extract(float* key, float* red_v,
                                                    int* red_i, int tid) {
  float bv = -__builtin_inff();
  int bi = T;
#pragma unroll
  for (int i = tid; i < T; i += NTHR) {
    float v = key[i];
    if (v > bv || (v == bv && i < bi)) { bv = v; bi = i; }
  }
  // wave32 butterfly reduction (lexicographic max on (value, -index))
#pragma unroll
  for (int m = 16; m >= 1; m >>= 1) {
    float ov = __shfl_xor(bv, m, 32);
    int   oi = __shfl_xor(bi, m, 32);
    if (ov > bv || (ov == bv && oi < bi)) { bv = ov; bi = oi; }
  }
  if ((tid & 31) == 0) { red_v[tid >> 5] = bv; red_i[tid >> 5] = bi; }
  __syncthreads();
  float fbv = red_v[0];
  int   fbi = red_i[0];
#pragma unroll
  for (int w = 1; w < NWAVES; ++w) {
    float ov = red_v[w]; int oi = red_i[w];
    if (ov > fbv || (ov == fbv && oi < fbi)) { fbv = ov; fbi = oi; }
  }
  if (tid == 0) key[fbi] = -__builtin_inff();  // remove from future rounds
  __syncthreads();
  return fbi;
}

// ---------------------------------------------------------------------------
// Fused kernel: one workgroup per batch element.
//  Phase 1: top-k selections over scores (22 argmax extractions; the k-loops
//           are deliberately NOT unrolled to keep I-cache footprint small --
//           they are barrier-bound, unrolling only bloats code).
//  Phase 2: flow-based sub-selection among the 10 hard samples (thread 0).
//  Phase 3: gather the 16 selected 4KB feature rows to the output via the
//           CDNA5 async global->LDS->global path (ASYNCcnt, VGPR-bypass):
//           16 outstanding async loads pipelined before a single wait, then
//           16 async stores. Fully unrolled so the wave queues them
//           back-to-back.
//
// Output layout (floats), concatenated in reference return order:
//   [0,     B*1*F)  hard_feat_nor      row b
//   [B*1*F, B*4*F)  hard_feat_abn      row b*3 + j
//   [B*4*F, B*6*F)  conf_rgb_feat_nor  row b*2 + j
//   [B*6*F, B*16*F) conf_rgb_feat_abn  row b*10 + j
// ---------------------------------------------------------------------------
__global__ __launch_bounds__(NTHR) void hard_sample_fused_kernel(
    const float* __restrict__ feat,
    const float* __restrict__ scores_rgb,
    const float* __restrict__ scores_flow,
    float* __restrict__ out) {
  const int b   = blockIdx.x;
  const int tid = threadIdx.x;

  __shared__ float s_rgb[T];
  __shared__ float s_key[T];
  __shared__ float red_v[NWAVES];
  __shared__ int   red_i[NWAVES];
  __shared__ int   s_rows[SLOTS];
  __shared__ __align__(16) float tile[SLOTS * F];   // 64 KB staging tile

  const float* rgb = scores_rgb + (size_t)b * T;
#pragma unroll
  for (int i = tid; i < T; i += NTHR) s_rgb[i] = rgb[i];
  __syncthreads();

  // --- hard samples: 10 smallest |rgb - 0.5|  (top_k of -|rgb-0.5|) ---
#pragma unroll
  for (int i = tid; i < T; i += NTHR) s_key[i] = -fabsf(s_rgb[i] - 0.5f);
  __syncthreads();
  int hard[HARD_K];
#pragma unroll 1
  for (int k = 0; k < HARD_K; ++k)
    hard[k] = block_argmax_extract(s_key, red_v, red_i, tid);

  // --- confident normal: 2 smallest rgb (top_k of -rgb) ---
#pragma unroll
  for (int i = tid; i < T; i += NTHR) s_key[i] = -s_rgb[i];
  __syncthreads();
  int cnor[CONF_K_NOR];
#pragma unroll 1
  for (int k = 0; k < CONF_K_NOR; ++k)
    cnor[k] = block_argmax_extract(s_key, red_v, red_i, tid);

  // --- confident abnormal: 10 largest rgb ---
#pragma unroll
  for (int i = tid; i < T; i += NTHR) s_key[i] = s_rgb[i];
  __syncthreads();
  int cabn[CONF_K_ABN];
#pragma unroll 1
  for (int k = 0; k < CONF_K_ABN; ++k)
    cabn[k] = block_argmax_extract(s_key, red_v, red_i, tid);

  // --- flow-based sub-selection among the 10 hard samples (tiny; thread 0) ---
  if (tid == 0) {
    const float* flow = scores_flow + (size_t)b * T;
    float hf[HARD_K];
#pragma unroll 1
    for (int k = 0; k < HARD_K; ++k) hf[k] = flow[hard[k]];

    bool taken[HARD_K] = {};
    int abn[HARD_K_ABN];
#pragma unroll 1
    for (int j = 0; j < HARD_K_ABN; ++j) {       // top-3 flow, ties -> low idx
      float bvv = -__builtin_inff(); int bii = 0;
      for (int k = 0; k < HARD_K; ++k)
        if (!taken[k] && hf[k] > bvv) { bvv = hf[k]; bii = k; }
      taken[bii] = true;
      abn[j] = bii;
    }
    float nv = hf[0]; int ni = 0;                // bottom-1 flow, ties -> low idx
#pragma unroll 1
    for (int k = 1; k < HARD_K; ++k)
      if (hf[k] < nv) { nv = hf[k]; ni = k; }

    s_rows[0] = hard[ni];
    for (int j = 0; j < HARD_K_ABN; ++j) s_rows[1 + j] = hard[abn[j]];
    for (int j = 0; j < CONF_K_NOR; ++j) s_rows[4 + j] = cnor[j];
    for (int j = 0; j < CONF_K_ABN; ++j) s_rows[6 + j] = cabn[j];
  }
  __syncthreads();

  // --- Phase 3: async gather of 16 rows (each lane owns 16B per row) ---
  // Issue all 16 async global->LDS loads back-to-back (ASYNCcnt = 16
  // outstanding), one wait, then all 16 LDS->global stores.
#pragma unroll
  for (int slot = 0; slot < SLOTS; ++slot) {
    const float* src =
        feat + ((size_t)b * T + (size_t)s_rows[slot]) * F + tid * 4;
    // Low 32 bits of a generic pointer to LDS are the wave-relative LDS byte
    // address (ISA 10.2) -- exactly what the async ops' LDS-address VGPR wants.
    unsigned lds = (unsigned)(uintptr_t)(&tile[slot * F + tid * 4]);
    asm volatile("global_load_async_to_lds_b128 %0, %1, off"
                 :: "v"(lds), "v"(src) : "memory");
  }
  asm volatile("s_wait_asynccnt 0x0" ::: "memory");

#pragma unroll
  for (int slot = 0; slot < SLOTS; ++slot) {
    size_t dstRow;
    if (slot == 0)      dstRow = (size_t)b;
    else if (slot < 4)  dstRow = (size_t)B * 1 + (size_t)b * 3 + (size_t)(slot - 1);
    else if (slot < 6)  dstRow = (size_t)B * 4 + (size_t)b * 2 + (size_t)(slot - 4);
    else                dstRow = (size_t)B * 6 + (size_t)b * 10 + (size_t)(slot - 6);
    float* dst = out + dstRow * F + tid * 4;
    unsigned lds = (unsigned)(uintptr_t)(&tile[slot * F + tid * 4]);
    asm volatile("global_store_async_from_lds_b128 %0, %1, off"
                 :: "v"(dst), "v"(lds) : "memory");
  }
  asm volatile("s_wait_asynccnt 0x0" ::: "memory");
}

// ---------------------------------------------------------------------------
extern "C" void kernel_launch(void* const* d_in, const int* in_sizes, int n_in,
                              void* d_out, int out_size, void* d_ws,
                              size_t ws_size, hipStream_t stream) {
  (void)in_sizes; (void)n_in; (void)out_size; (void)d_ws; (void)ws_size;
  const float* feat = (const float*)d_in[0];   // [B, T, F] f32
  const float* rgb  = (const float*)d_in[1];   // [B, T]    f32
  const float* flow = (const float*)d_in[2];   // [B, T]    f32
  float* out = (float*)d_out;                  // B*16*F f32, concatenated

  hard_sample_fused_kernel<<<B, NTHR, 0, stream>>>(feat, rgb, flow, out);
}